// FLAOperator_28295244546754
// MI455X (gfx1250) — compile-verified
//
#include <hip/hip_runtime.h>
#include <stdint.h>

typedef float v2f __attribute__((ext_vector_type(2)));
typedef float v8f __attribute__((ext_vector_type(8)));

#define BHCNT   64      // B*H
#define NSEQ    4096
#define DDIM    128
#define CHUNKS  8
#define CLEN    512     // NSEQ / CHUNKS
#define TILE    64
#define NTILES  8       // CLEN / TILE
#define BLOCK   256

// LDS layout (float offsets). Tiles: 64 t x 128 d per array, 3 arrays, x2 buffers.
#define LQ(b)   ((b)*24576 + 0)
#define LK(b)   ((b)*24576 + 8192)
#define LV(b)   ((b)*24576 + 16384)
#define LALPHA  49152   // 64
#define LW      49216   // 64
#define LRD     49280   // 64
#define LSC     49344   // 1
#define LHA     49352   // 128 (first-half inclusive totals per channel)
#define LLT     49480   // 128 (second-half local totals per channel)
#define LDS_FLOATS 49608
#define SUM_STRIDE 130  // n[128], d, m

// Summaries at ws[0 .. 512*130), exclusive prefixes at ws[512*130 .. 2*512*130)

template <bool WRITE_Y>
__global__ __launch_bounds__(BLOCK, 1) void fla_scan_kernel(
    const float* __restrict__ qg, const float* __restrict__ kg,
    const float* __restrict__ vg, float* __restrict__ ws,
    float* __restrict__ yg) {
  extern __shared__ float smem[];
  const int tid  = threadIdx.x;
  const int lane = tid & 31;
  const int wave = tid >> 5;
  const int wg   = blockIdx.x;
  const int bh   = wg >> 3;   // / CHUNKS
  const int cch  = wg & 7;    // chunk id

  const unsigned ldsbase = (unsigned)(size_t)(void*)smem;

  // Persistent carries. m/d live meaningfully in wave 0; acc[d] in tid<128.
  float m_carry = -INFINITY;
  float d_carry = 0.0f;
  float acc     = 0.0f;

  if (WRITE_Y) {
    const float* pref = ws + (size_t)512 * SUM_STRIDE + (size_t)wg * SUM_STRIDE;
    if (tid < 128) acc = pref[tid];
    d_carry = pref[128];
    m_carry = pref[129];
  }

  auto issue_tile_loads = [&](int tile, int buf) {
    const size_t rowbase =
        (size_t)bh * NSEQ + (size_t)cch * CLEN + (size_t)tile * TILE;
#pragma unroll
    for (int j = 0; j < 24; ++j) {
      const int idx = wave + (j << 3);   // 0..191 spread across 8 waves
      const int arr = idx >> 6;          // 0=q 1=k 2=v
      const int row = idx & 63;
      const float* g = (arr == 0) ? qg : (arr == 1) ? kg : vg;
      const uint64_t ga =
          (uint64_t)(const void*)(g + (rowbase + row) * DDIM + (lane << 2));
      const unsigned la =
          ldsbase +
          (unsigned)((buf * 24576 + arr * 8192 + row * 128 + (lane << 2)) << 2);
      asm volatile("global_load_async_to_lds_b128 %0, %1, off" ::"v"(la),
                       "v"(ga)
                   : "memory");
    }
  };

  issue_tile_loads(0, 0);

  for (int t = 0; t < NTILES; ++t) {
    const int buf = t & 1;
    if (t + 1 < NTILES) {
      issue_tile_loads(t + 1, buf ^ 1);
      asm volatile("s_wait_asynccnt 24" ::: "memory");  // tile t complete
    } else {
      asm volatile("s_wait_asynccnt 0" ::: "memory");
    }
    __syncthreads();

    // ---- Phase A: alpha_t = q_t . k_t via WMMA diag of Q K^T (waves 0..3) ----
    if (wave < 4) {
      const int r  = lane & 15;           // matrix row / col index
      const int kb = (lane >> 4) << 1;    // K sub-lane offset: 0 or 2
      const float* qrow = &smem[LQ(buf) + (wave * 16 + r) * 128 + kb];
      const float* krow = &smem[LK(buf) + (wave * 16 + r) * 128 + kb];
      v8f cacc = {};
#pragma unroll
      for (int k0 = 0; k0 < 128; k0 += 4) {
        v2f a = *(const v2f*)(qrow + k0);  // A[16x4] slice
        v2f b = *(const v2f*)(krow + k0);  // B[4x16] slice (K row-major == B^T)
        cacc = __builtin_amdgcn_wmma_f32_16x16x4_f32(false, a, false, b,
                                                     (short)0, cacc, false,
                                                     false);
      }
      // Diagonal: m<8 -> (vgpr m, lane m); m>=8 -> (vgpr m-8, lane m+16)
      float dv = cacc[0];
      dv = ((lane & 7) == 1) ? cacc[1] : dv;
      dv = ((lane & 7) == 2) ? cacc[2] : dv;
      dv = ((lane & 7) == 3) ? cacc[3] : dv;
      dv = ((lane & 7) == 4) ? cacc[4] : dv;
      dv = ((lane & 7) == 5) ? cacc[5] : dv;
      dv = ((lane & 7) == 6) ? cacc[6] : dv;
      dv = ((lane & 7) == 7) ? cacc[7] : dv;
      if (lane < 8 || lane >= 24) {
        const int m = (lane < 8) ? lane : (lane - 16);
        smem[LALPHA + wave * 16 + m] = dv;
      }
    }
    __syncthreads();

    // ---- Phase B: tile max, weights, (prefix) sums — wave 0 only ----
    if (wave == 0) {
      float a0 = smem[LALPHA + lane];
      float a1 = smem[LALPHA + 32 + lane];
      float mx = fmaxf(a0, a1);
#pragma unroll
      for (int s = 16; s >= 1; s >>= 1) mx = fmaxf(mx, __shfl_xor(mx, s));
      const float Mn = fmaxf(m_carry, mx);
      const float sc = __expf(m_carry - Mn);
      const float w0 = __expf(a0 - Mn);
      const float w1 = __expf(a1 - Mn);
      smem[LW + lane]      = w0;
      smem[LW + 32 + lane] = w1;
      if (WRITE_Y) {
        float s0 = w0;
#pragma unroll
        for (int d2 = 1; d2 < 32; d2 <<= 1) {
          float tt = __shfl_up(s0, d2);
          if (lane >= d2) s0 += tt;
        }
        const float tot0 = __shfl(s0, 31);
        float s1 = w1;
#pragma unroll
        for (int d2 = 1; d2 < 32; d2 <<= 1) {
          float tt = __shfl_up(s1, d2);
          if (lane >= d2) s1 += tt;
        }
        s1 += tot0;
        const float dbase = d_carry * sc;
        smem[LRD + lane]      = 1.0f / (dbase + s0);
        smem[LRD + 32 + lane] = 1.0f / (dbase + s1);
        d_carry = dbase + __shfl(s1, 31);
      } else {
        float tw = w0 + w1;
#pragma unroll
        for (int s = 16; s >= 1; s >>= 1) tw += __shfl_xor(tw, s);
        d_carry = d_carry * sc + tw;
      }
      m_carry = Mn;
      if (lane == 0) smem[LSC] = sc;
    }
    __syncthreads();

    // ---- Phase C: channel-parallel vector prefix over t (all 256 threads) ----
    const float scb = smem[LSC];
    const int ch   = tid & 127;
    float afirst = 0.0f;
    float pre[32];
    if (tid < 128) {
      float a = acc * scb;
      if (WRITE_Y) {
        float* yout =
            yg + ((size_t)bh * NSEQ + (size_t)cch * CLEN + (size_t)t * TILE) *
                     DDIM +
            ch;
#pragma unroll
        for (int s = 0; s < 32; ++s) {
          a += smem[LW + s] * smem[LV(buf) + s * 128 + ch];
          yout[(size_t)s * DDIM] = a * smem[LRD + s];
        }
      } else {
#pragma unroll
        for (int s = 0; s < 32; ++s)
          a += smem[LW + s] * smem[LV(buf) + s * 128 + ch];
      }
      smem[LHA + ch] = a;
      afirst = a;
    } else {
      float p = 0.0f;
#pragma unroll
      for (int s = 0; s < 32; ++s) {
        p += smem[LW + 32 + s] * smem[LV(buf) + (32 + s) * 128 + ch];
        pre[s] = p;
      }
      smem[LLT + ch] = p;
    }
    __syncthreads();
    if (tid < 128) {
      acc = afirst + smem[LLT + ch];
    } else if (WRITE_Y) {
      const float base = smem[LHA + ch];
      float* yout = yg +
                    ((size_t)bh * NSEQ + (size_t)cch * CLEN +
                     (size_t)t * TILE + 32) *
                        DDIM +
                    ch;
#pragma unroll
      for (int s = 0; s < 32; ++s)
        yout[(size_t)s * DDIM] = (base + pre[s]) * smem[LRD + 32 + s];
    }
    __syncthreads();
  }

  if (!WRITE_Y) {
    float* sums = ws + (size_t)wg * SUM_STRIDE;
    if (tid < 128) sums[tid] = acc;
    if (tid == 0) {
      sums[128] = d_carry;
      sums[129] = m_carry;
    }
  }
}

// Exclusive combine of chunk summaries along each head (64 blocks x 128 thr).
__global__ __launch_bounds__(128, 1) void fla_combine_kernel(float* ws) {
  const int bh = blockIdx.x;
  const int d  = threadIdx.x;
  float* sums  = ws;
  float* prefs = ws + (size_t)512 * SUM_STRIDE;
  float n_p = 0.0f, d_p = 0.0f, m_p = -INFINITY;
  for (int c = 0; c < CHUNKS; ++c) {
    const size_t idx = (size_t)(bh * CHUNKS + c) * SUM_STRIDE;
    prefs[idx + d] = n_p;
    if (d == 0) {
      prefs[idx + 128] = d_p;
      prefs[idx + 129] = m_p;
    }
    const float sn = sums[idx + d];
    const float sd = sums[idx + 128];
    const float sm = sums[idx + 129];
    const float M  = fmaxf(m_p, sm);
    const float e0 = __expf(m_p - M);
    const float e1 = __expf(sm - M);
    n_p = e0 * n_p + e1 * sn;
    d_p = e0 * d_p + e1 * sd;
    m_p = M;
  }
}

extern "C" void kernel_launch(void* const* d_in, const int* in_sizes, int n_in,
                              void* d_out, int out_size, void* d_ws,
                              size_t ws_size, hipStream_t stream) {
  const float* q = (const float*)d_in[0];
  const float* k = (const float*)d_in[1];
  const float* v = (const float*)d_in[2];
  float* y  = (float*)d_out;
  float* ws = (float*)d_ws;

  const size_t shmem = (size_t)LDS_FLOATS * sizeof(float);  // ~194 KB of 320 KB
  hipFuncSetAttribute(reinterpret_cast<const void*>(&fla_scan_kernel<false>),
                      hipFuncAttributeMaxDynamicSharedMemorySize, (int)shmem);
  hipFuncSetAttribute(reinterpret_cast<const void*>(&fla_scan_kernel<true>),
                      hipFuncAttributeMaxDynamicSharedMemorySize, (int)shmem);

  // Pass 1: per-chunk local scans -> summaries (n[128], d, m)
  fla_scan_kernel<false>
      <<<BHCNT * CHUNKS, BLOCK, shmem, stream>>>(q, k, v, ws, nullptr);
  // Pass 2: exclusive prefix combine across chunks of each head
  fla_combine_kernel<<<BHCNT, 128, 0, stream>>>(ws);
  // Pass 3: seeded rescan writing y
  fla_scan_kernel<true>
      <<<BHCNT * CHUNKS, BLOCK, shmem, stream>>>(q, k, v, ws, y);
}